// RejectionSampler_85023172591626
// MI455X (gfx1250) — compile-verified
//
#include <hip/hip_runtime.h>
#include <hip/hip_bf16.h>
#include <float.h>

#ifndef __has_builtin
#define __has_builtin(x) 0
#endif

// CDNA5 async global->LDS path (ASYNCcnt). Guarded so the file always compiles;
// falls back to nontemporal b128 loads if the builtins are absent.
#if defined(__HIP_DEVICE_COMPILE__) && \
    __has_builtin(__builtin_amdgcn_global_load_async_to_lds_b128) && \
    __has_builtin(__builtin_amdgcn_s_wait_asynccnt)
#define USE_ASYNC_LDS 1
#else
#define USE_ASYNC_LDS 0
#endif

#define TPB   256                 // 8 wave32 waves per workgroup
#define DEPTH 4                   // async pipeline depth (4 LDS buffers)
#define CHUNK (TPB * 4)           // 1024 floats = 4 KB per chunk

typedef __attribute__((ext_vector_type(4))) float f4;

// Types matching the async-load builtin's parameters:
//   param0: int4 (vector_size 16) in addrspace(1)   param1: int4 in addrspace(3)
typedef int v4i __attribute__((vector_size(16)));
typedef __attribute__((address_space(1))) v4i gv4i;
typedef __attribute__((address_space(3))) v4i lv4i;

// ---------------------------------------------------------------------------
// Kernel 1: per-row argmax over [T, VOCAB] float32 (first-index-wins ties).
// One workgroup per row; streams the row through LDS via async DMA.
// ---------------------------------------------------------------------------
__global__ __launch_bounds__(TPB) void argmax_rows_kernel(
    const float* __restrict__ logits, int vocab, int* __restrict__ row_argmax)
{
    const int row = blockIdx.x;
    const int tid = threadIdx.x;
    const float* __restrict__ rowp = logits + (size_t)row * (size_t)vocab;

    float best    = -FLT_MAX;
    int   bestIdx = 0x7fffffff;

    const int nchunks = vocab / CHUNK;

#if USE_ASYNC_LDS
    __shared__ float buf[DEPTH][CHUNK];

    auto issue = [&](int c) {
        const float* g = rowp + (size_t)c * CHUNK + (size_t)tid * 4;
        __builtin_amdgcn_global_load_async_to_lds_b128(
            (gv4i*)g, (lv4i*)&buf[c & (DEPTH - 1)][tid * 4], 0, 0);
    };

    const int head = (nchunks < DEPTH) ? nchunks : DEPTH;
    for (int p = 0; p < head; ++p) issue(p);

    for (int c = 0; c < nchunks; ++c) {
        const int rem = nchunks - 1 - c;   // loads allowed to stay outstanding
        if (rem >= DEPTH - 1)  __builtin_amdgcn_s_wait_asynccnt(DEPTH - 1);
        else if (rem == 2)     __builtin_amdgcn_s_wait_asynccnt(2);
        else if (rem == 1)     __builtin_amdgcn_s_wait_asynccnt(1);
        else                   __builtin_amdgcn_s_wait_asynccnt(0);
        asm volatile("" ::: "memory");     // keep LDS reads after the wait

        const f4  v    = *(const f4*)&buf[c & (DEPTH - 1)][tid * 4];
        const int base = c * CHUNK + tid * 4;
        if (v.x > best) { best = v.x; bestIdx = base;     }
        if (v.y > best) { best = v.y; bestIdx = base + 1; }
        if (v.z > best) { best = v.z; bestIdx = base + 2; }
        if (v.w > best) { best = v.w; bestIdx = base + 3; }

        asm volatile("" ::: "memory");     // consume before re-issuing the slot
        if (c + DEPTH < nchunks) issue(c + DEPTH);
    }
#else
    for (int c = 0; c < nchunks; ++c) {
        const int base = c * CHUNK + tid * 4;
        const f4  v    = __builtin_nontemporal_load((const f4*)(rowp + base));
        if (v.x > best) { best = v.x; bestIdx = base;     }
        if (v.y > best) { best = v.y; bestIdx = base + 1; }
        if (v.z > best) { best = v.z; bestIdx = base + 2; }
        if (v.w > best) { best = v.w; bestIdx = base + 3; }
    }
#endif

    // Remainder (indices larger than all processed ones -> strict > is enough)
    for (int i = nchunks * CHUNK + tid; i < vocab; i += TPB) {
        const float v = rowp[i];
        if (v > best) { best = v; bestIdx = i; }
    }

    // wave32 butterfly reduction, first-index-wins on ties
    #pragma unroll
    for (int off = 16; off > 0; off >>= 1) {
        const float ov = __shfl_xor(best, off, 32);
        const int   oi = __shfl_xor(bestIdx, off, 32);
        if (ov > best || (ov == best && oi < bestIdx)) { best = ov; bestIdx = oi; }
    }

    __shared__ float swv[TPB / 32];
    __shared__ int   swi[TPB / 32];
    const int wave = tid >> 5;
    if ((tid & 31) == 0) { swv[wave] = best; swi[wave] = bestIdx; }
    __syncthreads();
    if (tid == 0) {
        float bv = swv[0]; int bi = swi[0];
        #pragma unroll
        for (int w = 1; w < TPB / 32; ++w) {
            if (swv[w] > bv || (swv[w] == bv && swi[w] < bi)) { bv = swv[w]; bi = swi[w]; }
        }
        row_argmax[row] = bi;
    }
}

// ---------------------------------------------------------------------------
// Kernel 2: rejection bookkeeping, one thread per batch element.
// Writes output_token_ids [B, S+1] then num_bonus_tokens [B] (as float).
// ---------------------------------------------------------------------------
__global__ void finalize_kernel(const int* __restrict__ draft,
                                const int* __restrict__ cu,
                                const int* __restrict__ bonus,
                                const int* __restrict__ row_argmax,
                                int B, int S, float* __restrict__ out)
{
    const int b = blockIdx.x * blockDim.x + threadIdx.x;
    if (b >= B) return;

    const int start = (b == 0) ? 0 : cu[b - 1];
    int n = cu[b] - start;
    if (n > S) n = S;

    float* rowo = out + (size_t)b * (size_t)(S + 1);
    bool rejected  = false;
    int  processed = 0;

    for (int j = 0; j <= S; ++j) {
        int v;
        if (j < n) {
            if (!rejected) {
                const int t = row_argmax[start + j];
                v = t;                              // argmax stored at mismatch pos itself
                ++processed;
                if (draft[start + j] != t) rejected = true;
            } else {
                v = -1;                             // positions after first mismatch
            }
        } else if (j == n) {
            v = rejected ? -1 : bonus[b];           // bonus slot
        } else {
            v = -1;                                 // past bonus
        }
        rowo[j] = (float)v;
    }
    out[(size_t)B * (size_t)(S + 1) + b] =
        (float)(processed - 1 + (rejected ? 0 : 1));
}

// ---------------------------------------------------------------------------
// Host launch. Inputs (setup_inputs order):
//   0: draft_token_ids [T] i32   1: num_spec_steps (scalar)   2: cu [B] i32
//   3: target_logits [T*V] f32   4: bonus_token_ids [B] i32
// ---------------------------------------------------------------------------
extern "C" void kernel_launch(void* const* d_in, const int* in_sizes, int n_in,
                              void* d_out, int out_size, void* d_ws, size_t ws_size,
                              hipStream_t stream) {
    const int*   draft  = (const int*)d_in[0];
    const int*   cu     = (const int*)d_in[2];
    const float* logits = (const float*)d_in[3];
    const int*   bonus  = (const int*)d_in[4];

    const int T     = in_sizes[0];
    const int B     = in_sizes[2];
    const int vocab = in_sizes[3] / T;
    const int S     = out_size / B - 2;   // out = B*(S+1) tokens + B counts

    int* row_argmax = (int*)d_ws;         // T ints of scratch

    argmax_rows_kernel<<<T, TPB, 0, stream>>>(logits, vocab, row_argmax);
    finalize_kernel<<<(B + 127) / 128, 128, 0, stream>>>(
        draft, cu, bonus, row_argmax, B, S, (float*)d_out);
}